// CTF_75436805587557
// MI455X (gfx1250) — compile-verified
//
#include <hip/hip_runtime.h>
#include <hip/hip_bf16.h>
#include <math.h>

// ---------------------------------------------------------------------------
// CTF forward: per (b,f) solve min || A x - y ||  with A [256 x 192] complex
// built from 3 banded Toeplitz blocks of normalized X_hat.
// Normal equations G = A^H A (Hermitian 192x192, packed lower in LDS):
//   - Gram matrix via V_WMMA_F32_16X16X4_F32 (branch-free guarded loads)
//   - blocked Cholesky: panel width 16, rank-16 trailing update ALSO on WMMA
//   - forward/backward substitution on a single wave (in-order LDS, no
//     workgroup barriers)
// ---------------------------------------------------------------------------

typedef __attribute__((ext_vector_type(2))) float v2f;
typedef __attribute__((ext_vector_type(8))) float v8f;

#define CTF_B      4
#define CTF_F      257
#define CTF_N      256
#define CTF_NK     64
#define CTF_BANDS  3
#define CTF_COLS   192            // BANDS * NK
#define CTF_TILES  78             // 12*13/2 lower-triangle 16x16 tiles
#define NBLK       12             // 192 / 16 panels
#define PACKED     18528          // 192*193/2
#define VPAD       320            // 64 zero guard + 256 data per band
#define EPS_LAMBDA 1.0f
#define PAD_VAL    1e-8f

// LDS layout (floats)
#define L_VR   0                        // 3*320 (guarded)
#define L_VI   (L_VR + 3*VPAD)          // 3*320
#define L_YR   (L_VI + 3*VPAD)          // 256
#define L_YI   (L_YR + 256)             // 256
#define L_RR   (L_YI + 256)             // 192
#define L_RI   (L_RR + 192)             // 192
#define L_CR   (L_RI + 192)             // 192 column cache (re)
#define L_CI   (L_CR + 192)             // 192 column cache (im)
#define L_GR   (L_CI + 192)             // 18528 packed lower
#define L_GI   (L_GR + PACKED)          // 18528 packed lower
#define L_SC   (L_GI + PACKED)          // scalars
#define LDS_FLOATS (L_SC + 2)           // 40258 floats = 161,032 bytes

// ---------------------------------------------------------------------------
__global__ void ctf_init_kernel(unsigned int* g) {
    if (threadIdx.x < CTF_B) g[threadIdx.x] = 0u;
}

// ---------------------------------------------------------------------------
// gmax[b] = max over (F,N) of |Y|^2 (atomicMax on float bits; valid for >=0).
// ---------------------------------------------------------------------------
__global__ __launch_bounds__(256) void ctf_max_kernel(const float* __restrict__ Yr,
                                                      const float* __restrict__ Yi,
                                                      unsigned int* __restrict__ gmax) {
    __shared__ float red[256];
    const int b = blockIdx.y;
    const unsigned base = (unsigned)b * (CTF_F * CTF_N);
    float m = 0.0f;
    for (unsigned idx = blockIdx.x * 256u + threadIdx.x;
         idx < (unsigned)(CTF_F * CTF_N);
         idx += gridDim.x * 256u) {
        float a = Yr[base + idx], c = Yi[base + idx];
        m = fmaxf(m, a * a + c * c);
    }
    red[threadIdx.x] = m;
    __syncthreads();
    for (int s = 128; s > 0; s >>= 1) {
        if ((int)threadIdx.x < s)
            red[threadIdx.x] = fmaxf(red[threadIdx.x], red[threadIdx.x + s]);
        __syncthreads();
    }
    if (threadIdx.x == 0) atomicMax(&gmax[b], __float_as_uint(red[0]));
}

// ---------------------------------------------------------------------------
// One workgroup (256 threads = 8 waves) per (b,f) problem.
// ---------------------------------------------------------------------------
__global__ __launch_bounds__(256) void ctf_main_kernel(const float* __restrict__ Yr,
                                                       const float* __restrict__ Yi,
                                                       const float* __restrict__ Xr,
                                                       const float* __restrict__ Xi,
                                                       const float* __restrict__ gmaxF,
                                                       float* __restrict__ out) {
    extern __shared__ float smem[];
    float* vr   = smem + L_VR;
    float* vi   = smem + L_VI;
    float* yr   = smem + L_YR;
    float* yi   = smem + L_YI;
    float* rr   = smem + L_RR;
    float* ri   = smem + L_RI;
    float* colr = smem + L_CR;
    float* coli = smem + L_CI;
    float* Grp  = smem + L_GR;
    float* Gip  = smem + L_GI;
    float* sc   = smem + L_SC;

    const int bf  = blockIdx.x;
    const int b   = bf / CTF_F;
    const int f   = bf % CTF_F;
    const int tid = threadIdx.x;

    const float gm = EPS_LAMBDA * gmaxF[b];
    const unsigned baseB = (unsigned)b * (CTF_F * CTF_N);

    // ---- stage normalized y and the 3 guarded band vectors into LDS ------
    if (tid < CTF_COLS) {                       // zero the 3x64 front guards
        const int band = tid >> 6, g = tid & 63;
        vr[band * VPAD + g] = 0.0f;
        vi[band * VPAD + g] = 0.0f;
    }
    {
        const int n = tid;                      // 256 threads == 256 frames
        unsigned o = baseB + (unsigned)f * CTF_N + n;
        float a = Yr[o], c = Yi[o];
        float s = sqrtf(fmaxf(gm, a * a + c * c));
        float inv = 1.0f / s;
        yr[n] = a * inv;
        yi[n] = c * inv;
        for (int band = 0; band < CTF_BANDS; ++band) {
            int fs = f + band - 1;              // F_PRIM = 1
            float wr, wi;
            if (fs < 0 || fs >= CTF_F) {
                wr = PAD_VAL; wi = 0.0f;        // pad applied AFTER normalization
            } else {
                unsigned o2 = baseB + (unsigned)fs * CTF_N + n;
                float a2 = Yr[o2], c2 = Yi[o2];
                float s2 = sqrtf(fmaxf(gm, a2 * a2 + c2 * c2));
                float inv2 = 1.0f / s2;
                wr = Xr[o2] * inv2;
                wi = Xi[o2] * inv2;
            }
            vr[band * VPAD + 64 + n] = wr;
            vi[band * VPAD + 64 + n] = wi;
        }
    }
    __syncthreads();

    // ---- rhs = A^H y  (192 complex dot products, VALU) -------------------
    if (tid < CTF_COLS) {
        const int band = tid >> 6, k = tid & 63;
        const float* ar = vr + band * VPAD + 64;
        const float* ai = vi + band * VPAD + 64;
        float sr = 0.0f, si = 0.0f;
        for (int m = k; m < CTF_N; ++m) {
            float xre = ar[m - k], xim = ai[m - k];
            float yre = yr[m],     yim = yi[m];
            sr += xre * yre + xim * yim;        // conj(A) * y
            si += xre * yim - xim * yre;
        }
        rr[tid] = sr;
        ri[tid] = si;
    }

    // ---- G = A^H A lower triangle via f32 WMMA ---------------------------
    // A-operand (16x4): lane l -> M = l&15, VGPR{0,1} hold K{0,1} (l<16) or
    // K{2,3} (l>=16). B-operand (4x16) mirrors it; one branch-free loader
    // serves both. D: lane l -> N=l&15, VGPR r -> M = r + 8*(l>>4).
    const int wave    = tid >> 5;
    const int lane    = tid & 31;
    const int colLane = lane & 15;
    const int mOff    = (lane >> 4) << 1;       // 0 or 2
    const int rowOff  = (lane >> 4) * 8;        // D-matrix M offset

    for (int t = wave; t < CTF_TILES; t += 8) {
        int P = 0;
        while ((P + 1) * (P + 2) / 2 <= t) ++P; // t -> (P,Q), Q <= P
        int Q = t - P * (P + 1) / 2;

        const int cp = P * 16 + colLane;        // G row index of this lane
        const int cq = Q * 16 + colLane;        // G col index of this lane
        const float* pAr = vr + (cp >> 6) * VPAD + 64 - (cp & 63);
        const float* pAi = vi + (cp >> 6) * VPAD + 64 - (cp & 63);
        const float* pBr = vr + (cq >> 6) * VPAD + 64 - (cq & 63);
        const float* pBi = vi + (cq >> 6) * VPAD + 64 - (cq & 63);

        v8f accR = {};                          // Ar^T Ar + Ai^T Ai
        v8f accP = {};                          // Ar^T Ai
        v8f accN = {};                          // Ai^T Ar
#pragma unroll 4
        for (int m0 = 0; m0 < CTF_N; m0 += 4) {
            const int m = m0 + mOff;
            v2f arp, aip, arq, aiq;
            arp.x = pAr[m];  arp.y = pAr[m + 1];
            aip.x = pAi[m];  aip.y = pAi[m + 1];
            arq.x = pBr[m];  arq.y = pBr[m + 1];
            aiq.x = pBi[m];  aiq.y = pBi[m + 1];

            accR = __builtin_amdgcn_wmma_f32_16x16x4_f32(
                false, arp, false, arq, (short)0, accR, false, false);
            accR = __builtin_amdgcn_wmma_f32_16x16x4_f32(
                false, aip, false, aiq, (short)0, accR, false, false);
            accP = __builtin_amdgcn_wmma_f32_16x16x4_f32(
                false, arp, false, aiq, (short)0, accP, false, false);
            accN = __builtin_amdgcn_wmma_f32_16x16x4_f32(
                false, aip, false, arq, (short)0, accN, false, false);
        }

        for (int r = 0; r < 8; ++r) {           // scatter into packed lower
            int gi = P * 16 + r + rowOff;
            int gj = Q * 16 + colLane;
            if (gj <= gi) {
                int idx = gi * (gi + 1) / 2 + gj;
                Grp[idx] = accR[r];
                Gip[idx] = accP[r] - accN[r];   // Gi = Ar^T Ai - Ai^T Ar
            }
        }
    }
    __syncthreads();

    // ---- blocked complex Cholesky: 16-wide panels ------------------------
    for (int p = 0; p < NBLK; ++p) {
        const int j0 = p * 16, j1 = j0 + 16;

        // -- factor panel columns (narrow rank-1 updates, width <= 15) -----
        for (int j = j0; j < j1; ++j) {
            if (tid == 0) {
                float d = sqrtf(fmaxf(Grp[j * (j + 1) / 2 + j], 1e-30f));
                Grp[j * (j + 1) / 2 + j] = d;
                sc[0] = 1.0f / d;
            }
            __syncthreads();
            const float invd = sc[0];
            for (int i = j + 1 + tid; i < CTF_COLS; i += 256) {
                int idx = i * (i + 1) / 2 + j;
                float lr = Grp[idx] * invd;
                float li = Gip[idx] * invd;
                Grp[idx] = lr;  Gip[idx] = li;
                colr[i]  = lr;  coli[i]  = li;  // linear column cache
            }
            __syncthreads();
            for (int i = j + 1 + tid; i < CTF_COLS; i += 256) {
                const int rowbase = i * (i + 1) / 2;
                const float lir = colr[i], lii = coli[i];
                const int kend = (i < j1 - 1) ? i : (j1 - 1);
                for (int k2 = j + 1; k2 <= kend; ++k2) {
                    const float lkr = colr[k2], lki = coli[k2];
                    // G[i][k2] -= L[i][j] * conj(L[k2][j])
                    Grp[rowbase + k2] -= lir * lkr + lii * lki;
                    Gip[rowbase + k2] -= lii * lkr - lir * lki;
                }
            }
            __syncthreads();
        }

        // -- rank-16 trailing update via WMMA: G -= Lp * Lp^H --------------
        const int np = NBLK - 1 - p;            // trailing block count
        const int ntiles = np * (np + 1) / 2;
        for (int t = wave; t < ntiles; t += 8) {
            int Pp = 0;
            while ((Pp + 1) * (Pp + 2) / 2 <= t) ++Pp;
            int Qp = t - Pp * (Pp + 1) / 2;
            const int P = p + 1 + Pp, Q = p + 1 + Qp;

            const int rI = P * 16 + colLane;    // L row for A operand (M dim)
            const int rK = Q * 16 + colLane;    // L row for B operand (N dim)
            const int abase = rI * (rI + 1) / 2 + j0;   // L[rI][j0+jj]
            const int bbase = rK * (rK + 1) / 2 + j0;   // L[rK][j0+jj]

            v8f accR = {};                      // Lr Lr^T + Li Li^T
            v8f accP = {};                      // Lr Li^T
            v8f accN = {};                      // Li Lr^T
#pragma unroll
            for (int jj0 = 0; jj0 < 16; jj0 += 4) {
                const int jj = jj0 + mOff;
                v2f lrA, liA, lrB, liB;
                lrA.x = Grp[abase + jj];  lrA.y = Grp[abase + jj + 1];
                liA.x = Gip[abase + jj];  liA.y = Gip[abase + jj + 1];
                lrB.x = Grp[bbase + jj];  lrB.y = Grp[bbase + jj + 1];
                liB.x = Gip[bbase + jj];  liB.y = Gip[bbase + jj + 1];

                accR = __builtin_amdgcn_wmma_f32_16x16x4_f32(
                    false, lrA, false, lrB, (short)0, accR, false, false);
                accR = __builtin_amdgcn_wmma_f32_16x16x4_f32(
                    false, liA, false, liB, (short)0, accR, false, false);
                accP = __builtin_amdgcn_wmma_f32_16x16x4_f32(
                    false, lrA, false, liB, (short)0, accP, false, false);
                accN = __builtin_amdgcn_wmma_f32_16x16x4_f32(
                    false, liA, false, lrB, (short)0, accN, false, false);
            }

            for (int r = 0; r < 8; ++r) {       // RMW packed lower storage
                int gi = P * 16 + r + rowOff;
                int gj = Q * 16 + colLane;
                if (gj <= gi) {
                    int idx = gi * (gi + 1) / 2 + gj;
                    Grp[idx] -= accR[r];
                    Gip[idx] -= accN[r] - accP[r];  // Gi -= Li Lr^T - Lr Li^T
                }
            }
        }
        __syncthreads();
    }

    // ---- triangular solves on a single wave (in-order LDS, no barriers) --
    if (wave == 0) {
        // forward: L z = rhs (in place in rr/ri)
        for (int j = 0; j < CTF_COLS; ++j) {
            const float invd = 1.0f / Grp[j * (j + 1) / 2 + j];
            const float zr = rr[j] * invd;
            const float zi = ri[j] * invd;
            if (lane == 0) { rr[j] = zr; ri[j] = zi; }
            for (int i = j + 1 + lane; i < CTF_COLS; i += 32) {
                int idx = i * (i + 1) / 2 + j;
                float lr = Grp[idx], li = Gip[idx];
                rr[i] -= lr * zr - li * zi;
                ri[i] -= lr * zi + li * zr;
            }
        }
        // backward: L^H x = z
        for (int j = CTF_COLS - 1; j >= 0; --j) {
            const float invd = 1.0f / Grp[j * (j + 1) / 2 + j];
            const float xr = rr[j] * invd;
            const float xi = ri[j] * invd;
            if (lane == 0) { rr[j] = xr; ri[j] = xi; }
            const int rowbase = j * (j + 1) / 2;
            for (int i = lane; i < j; i += 32) {
                float lr = Grp[rowbase + i], li = Gip[rowbase + i];
                // z[i] -= conj(L[j][i]) * x[j]
                rr[i] -= lr * xr + li * xi;
                ri[i] -= lr * xi - li * xr;
            }
        }
    }
    __syncthreads();

    // ---- emit complex64 output [B,F,NK,BANDS] ----------------------------
    if (tid < CTF_COLS) {
        const int band = tid >> 6, k = tid & 63;
        const unsigned o = (((unsigned)bf * CTF_NK) + k) * CTF_BANDS + band;
        out[2 * o]     = rr[tid];
        out[2 * o + 1] = ri[tid];
    }
}

// ---------------------------------------------------------------------------
extern "C" void kernel_launch(void* const* d_in, const int* in_sizes, int n_in,
                              void* d_out, int out_size, void* d_ws, size_t ws_size,
                              hipStream_t stream) {
    (void)in_sizes; (void)n_in; (void)out_size; (void)ws_size;
    const float* Yr = (const float*)d_in[0];
    const float* Yi = (const float*)d_in[1];
    const float* Xr = (const float*)d_in[2];
    const float* Xi = (const float*)d_in[3];
    float* out = (float*)d_out;
    unsigned int* gmaxU = (unsigned int*)d_ws;
    float* gmaxF = (float*)d_ws;

    ctf_init_kernel<<<1, 32, 0, stream>>>(gmaxU);

    dim3 gridMax(32, CTF_B);
    ctf_max_kernel<<<gridMax, 256, 0, stream>>>(Yr, Yi, gmaxU);

    const size_t shmem = (size_t)LDS_FLOATS * sizeof(float);   // ~161 KB
    ctf_main_kernel<<<CTF_B * CTF_F, 256, shmem, stream>>>(Yr, Yi, Xr, Xi,
                                                           gmaxF, out);
}